// ImprovedGNN_71244917506158
// MI455X (gfx1250) — compile-verified
//
#include <hip/hip_runtime.h>

#define N_NODES 50000
#define F_OUT   64
#define E_EDGES 800000

typedef __attribute__((ext_vector_type(2))) float v2f;
typedef __attribute__((ext_vector_type(8))) float v8f;

// ---------------------------------------------------------------- utilities

__global__ void zero_f32(float* __restrict__ p, int n) {
  int i = blockIdx.x * blockDim.x + threadIdx.x;
  int stride = gridDim.x * blockDim.x;
  for (; i < n; i += stride) p[i] = 0.0f;
}

__global__ void degree_accum(const int* __restrict__ src, const int* __restrict__ dst,
                             float* __restrict__ deg_out, float* __restrict__ deg_in, int E) {
  int e = blockIdx.x * blockDim.x + threadIdx.x;
  if (e < E) {
    atomicAdd(&deg_out[src[e]], 1.0f);
    atomicAdd(&deg_in[dst[e]], 1.0f);
  }
}

__global__ void isqrt_clamp(float* __restrict__ p, int n) {
  int i = blockIdx.x * blockDim.x + threadIdx.x;
  if (i < n) {
    float d = p[i];
    d = d < 1.0f ? 1.0f : d;
    p[i] = rsqrtf(d);
  }
}

// ------------------------------------------------- WMMA fp32 GEMM + row scale
// Out[N_NODES x 64] = (X[N_NODES x K] @ W[K x 64]) * rowScale[row]
// One wave32 computes one 16x16 tile via V_WMMA_F32_16X16X4_F32.
// A frag (16x4 f32): lanes 0-15 rows 0-15 with K={k0,k0+1}; lanes 16-31 K={k0+2,k0+3}.
// B frag (4x16 f32): mirrored (N striped across lanes, K-half by lane[4]).
// C/D (16x16 f32): vgpr r -> row r (lanes 0-15) / row r+8 (lanes 16-31), col = lane&15.
template <int K>
__global__ __launch_bounds__(256) void gemm_wmma_scale(
    const float* __restrict__ X, const float* __restrict__ W,
    const float* __restrict__ rowScale, float* __restrict__ Out) {
  const int lane = threadIdx.x & 31;
  const int wave = (blockIdx.x * blockDim.x + threadIdx.x) >> 5;
  const int numTiles = (N_NODES / 16) * (F_OUT / 16);
  if (wave >= numTiles) return;               // wave-uniform: EXEC all-1s in WMMA

  const int tileN = wave & (F_OUT / 16 - 1);  // 0..3
  const int tileM = wave >> 2;                // 0..3124
  const int r16 = lane & 15;
  const int kh  = lane >> 4;                  // 0 or 1: which K-half this lane holds
  const int row = tileM * 16 + r16;
  const int col = tileN * 16 + r16;

  const float* __restrict__ xrow = X + (long)row * K;

  v8f acc = {};
#pragma unroll 4
  for (int k0 = 0; k0 < K; k0 += 4) {
    const int kk = k0 + kh * 2;
    v2f a = *(const v2f*)(xrow + kk);          // A[row][kk], A[row][kk+1] (8B aligned)
    v2f b;
    b.x = W[kk * F_OUT + col];                 // B[kk][col]
    b.y = W[(kk + 1) * F_OUT + col];           // B[kk+1][col]
    acc = __builtin_amdgcn_wmma_f32_16x16x4_f32(
        /*neg_a=*/false, a, /*neg_b=*/false, b,
        /*c_mod=*/(short)0, acc, /*reuse_a=*/false, /*reuse_b=*/false);
  }

#pragma unroll
  for (int r = 0; r < 8; ++r) {
    const int m = tileM * 16 + r + kh * 8;
    Out[(long)m * F_OUT + col] = acc[r] * rowScale[m];
  }
}

// -------------------------------------------------- edge gather + scatter-add
// 16 threads per edge, float4 each: Agg[dst] += H[src]  (64 floats/edge)
__global__ void edge_scatter(const int* __restrict__ src, const int* __restrict__ dst,
                             const float* __restrict__ H, float* __restrict__ Agg, int E) {
  int t = blockIdx.x * blockDim.x + threadIdx.x;
  if (t >= E * 16) return;
  const int e = t >> 4;
  const int q = (t & 15) << 2;
  const int s = src[e];
  const int d = dst[e];
  const float4 v = *(const float4*)(H + (long)s * F_OUT + q);
  float* o = Agg + (long)d * F_OUT + q;
  atomicAdd(o + 0, v.x);
  atomicAdd(o + 1, v.y);
  atomicAdd(o + 2, v.z);
  atomicAdd(o + 3, v.w);
}

// ------------------------------------------------------- bias + scale (+relu)
__global__ void bias_act(const float* __restrict__ In, const float* __restrict__ dinIsqrt,
                         const float* __restrict__ bias, float* __restrict__ Out, int relu) {
  int i = blockIdx.x * blockDim.x + threadIdx.x;
  if (i >= N_NODES * F_OUT) return;
  const int n = i >> 6;
  const int f = i & 63;
  float v = In[i] * dinIsqrt[n] + bias[f];
  if (relu) v = v > 0.0f ? v : 0.0f;
  Out[i] = v;
}

// --------------------------------------------------------------------- driver

extern "C" void kernel_launch(void* const* d_in, const int* in_sizes, int n_in,
                              void* d_out, int out_size, void* d_ws, size_t ws_size,
                              hipStream_t stream) {
  (void)in_sizes; (void)n_in; (void)out_size; (void)ws_size;

  const float* x   = (const float*)d_in[0];   // [50000,128]
  const float* W1  = (const float*)d_in[1];   // [128,64]
  const float* b1  = (const float*)d_in[2];   // [64]
  const float* W2  = (const float*)d_in[3];   // [64,64]
  const float* b2  = (const float*)d_in[4];   // [64]
  const int*   src = (const int*)d_in[5];     // [800000]
  const int*   dst = (const int*)d_in[6];     // [800000]
  float*       out = (float*)d_out;           // [50000,64]

  // workspace layout (floats)
  float* ws   = (float*)d_ws;
  float* degO = ws;                 // 50048 (padded)
  float* degI = ws + 50048;         // 50048
  float* bufA = ws + 100096;        // 3,200,000
  float* bufB = bufA + (N_NODES * F_OUT);

  const int E = E_EDGES;
  const int NF = N_NODES * F_OUT;   // 3,200,000

  // 1) degrees -> isqrt (clamped at 1, matches DGL)
  zero_f32<<<1024, 256, 0, stream>>>(degO, 100096);                 // both arrays
  degree_accum<<<(E + 255) / 256, 256, 0, stream>>>(src, dst, degO, degI, E);
  isqrt_clamp<<<(100096 + 255) / 256, 256, 0, stream>>>(degO, 100096);

  // 2) layer 1 GEMM: bufA = (x @ W1) * degO_isqrt  (WMMA f32 16x16x4)
  {
    const int waves = (N_NODES / 16) * (F_OUT / 16);                // 12500
    const int blocks = (waves * 32 + 255) / 256;
    gemm_wmma_scale<128><<<blocks, 256, 0, stream>>>(x, W1, degO, bufA);
  }

  // 3) scatter-add over edges into bufB
  zero_f32<<<2048, 256, 0, stream>>>(bufB, NF);
  edge_scatter<<<(E * 16 + 255) / 256, 256, 0, stream>>>(src, dst, bufA, bufB, E);

  // 4) bufA = relu(bufB * degI_isqrt + b1)
  bias_act<<<(NF + 255) / 256, 256, 0, stream>>>(bufB, degI, b1, bufA, 1);

  // 5) layer 2 GEMM: bufB = (bufA @ W2) * degO_isqrt
  {
    const int waves = (N_NODES / 16) * (F_OUT / 16);
    const int blocks = (waves * 32 + 255) / 256;
    gemm_wmma_scale<64><<<blocks, 256, 0, stream>>>(bufA, W2, degO, bufB);
  }

  // 6) scatter-add into d_out, then final bias+scale in place (no relu)
  zero_f32<<<2048, 256, 0, stream>>>(out, NF);
  edge_scatter<<<(E * 16 + 255) / 256, 256, 0, stream>>>(src, dst, bufB, out, E);
  bias_act<<<(NF + 255) / 256, 256, 0, stream>>>(out, degI, b2, out, 0);
}